// DEERLIFNode_10668698763991
// MI455X (gfx1250) — compile-verified
//
#include <hip/hip_runtime.h>
#include <cstdint>
#include <cstddef>

// Problem constants (match reference: T=1024, B=32, F=512)
#define T_DIM   1024
#define BF      16384          // B*F
#define COLS    32             // columns handled per block (one per lane)
#define NWAVES  32             // waves per block (1024 threads, wave32)
#define CHUNK   32             // timesteps per wave (NWAVES*CHUNK == T_DIM)
#define LSTRIDE 33             // padded LDS row stride in floats (bank spread)

// ---------------------------------------------------------------------------
// CDNA5 async global->LDS copy (gfx1250): per-lane 4B gather into LDS,
// tracked by ASYNCcnt.  (CDNA5 ISA)
// ---------------------------------------------------------------------------
__device__ __forceinline__ void async_ld_b32_to_lds(uint32_t lds_byte_off,
                                                    uint64_t global_addr) {
  asm volatile("global_load_async_to_lds_b32 %0, %1, off"
               :: "v"(lds_byte_off), "v"(global_addr)
               : "memory");
}
__device__ __forceinline__ void wait_asynccnt0() {
  asm volatile("s_wait_asynccnt 0" ::: "memory");
}

// a = 0.5 * (1 - sg) with sg = 4*s*(1-s), s = sigmoid(4*(hc-VTH))
//   = 0.5 * tanh^2( 2*(hc-VTH) )          (single v_tanh_f32 on gfx1250)
__device__ __forceinline__ float scan_coeff(float hc) {
  const float w2 = 2.0f * (hc - 0.7f);
#if defined(__has_builtin) && __has_builtin(__builtin_amdgcn_tanhf)
  const float th = __builtin_amdgcn_tanhf(w2);
#else
  // tanh(w2) = (1 - e) / (1 + e),  e = exp(-2*w2) = exp2(-2*w2*log2(e))
  const float e  = __builtin_amdgcn_exp2f(w2 * (-2.0f * 1.4426950408889634f));
  const float th = (1.0f - e) * __builtin_amdgcn_rcpf(1.0f + e);
#endif
  return 0.5f * th * th;
}

// ---------------------------------------------------------------------------
// One DEER fixed-point iteration for this block's 32 columns.
//   pass1: per-wave, compute a[t], b[t] for its 32-step chunk (registers),
//          consuming last iteration's y from y_reg (+1 LDS boundary value),
//          compose chunk affine summary (A,B) -> LDS
//   scan : fold preceding chunk summaries to get y_start for this wave
//   pass2: y[t] = a[t]*y[t-1] + b[t] -> y_reg (+ boundary to LDS;
//          global spike/y writes fused into the LAST iteration)
// ---------------------------------------------------------------------------
template <bool FIRST, bool LAST>
__device__ __forceinline__ void lif_iteration(
    const float* xs,        // LDS: x tile [T_DIM][LSTRIDE]
    float2*      summ,      // LDS: chunk summaries [NWAVES][COLS]
    float*       ybound,    // LDS: per-chunk last y of prev iter [NWAVES][COLS]
    float (&y_reg)[CHUNK],  // this wave's y chunk from the previous iteration
    int wave, int lane, float v, float h0, float h1,
    float* out_spike, float* out_y, int col)
{
  const int t0 = wave * CHUNK;
  float a_reg[CHUNK];
  float b_reg[CHUNK];
  float A  = 1.0f;   // chunk-composed multiplier
  float Bc = 0.0f;   // chunk-composed offset

  float ys0 = v;
  if (!FIRST && wave != 0) ys0 = ybound[(wave - 1) * COLS + lane];

#pragma unroll
  for (int i = 0; i < CHUNK; ++i) {
    const int t = t0 + i;
    float ys;
    if (FIRST) {
      // warmstart guess shifted: [v_init, h0, h1, 0, 0, ...]
      ys = (t == 0) ? v : (t == 1) ? h0 : (t == 2) ? h1 : 0.0f;
    } else {
      ys = (i == 0) ? ys0 : y_reg[i - 1];   // y_old[t-1], consumed in order
    }
    const float xv = xs[t * LSTRIDE + lane];
    const float h  = 0.5f * (xv + ys);                 // ys + (x-ys)/TAU
    const float hc = fminf(fmaxf(h, -20.0f), 20.0f);
    const float a  = scan_coeff(hc);                   // -G
    const float b  = h - a * ys;                       // h + G*ys
    a_reg[i] = a;
    b_reg[i] = b;
    A  = a * A;
    Bc = fmaf(a, Bc, b);
  }

  summ[wave * COLS + lane] = make_float2(A, Bc);
  __syncthreads();

  // exclusive prefix over preceding chunks (uniform trip count per wave)
  float y = v;
  for (int k = 0; k < wave; ++k) {
    const float2 s2 = summ[k * COLS + lane];
    y = fmaf(s2.x, y, s2.y);
  }

#pragma unroll
  for (int i = 0; i < CHUNK; ++i) {
    y = fmaf(a_reg[i], y, b_reg[i]);
    y_reg[i] = y;
    if (LAST) {
      const int    t = t0 + i;
      const size_t o = (size_t)t * BF + (size_t)col;
      out_y[o]     = y;
      out_spike[o] = (y >= 0.7f) ? 1.0f : 0.0f;
    }
  }
  if (!LAST) {
    ybound[wave * COLS + lane] = y;   // chunk-last y for wave+1's next pass1
    __syncthreads();
  }
}

// ---------------------------------------------------------------------------
__global__ void __launch_bounds__(1024, 1)
deer_lif_kernel(const float* __restrict__ x,
                const float* __restrict__ v_init,
                float* __restrict__ out)
{
  extern __shared__ float smem[];
  float* xs = smem;                         // [T_DIM][LSTRIDE] x tile
  __shared__ float2 summ[NWAVES * COLS];    // 8 KB
  __shared__ float  ybound[NWAVES * COLS];  // 4 KB

  const int lane    = threadIdx.x & 31;
  const int wave    = (int)threadIdx.x >> 5;
  const int colbase = (int)blockIdx.x * COLS;
  const int col     = colbase + lane;

  // ---- async-stage this block's x tile (1024 x 32 f32) into LDS ----------
  const int t0 = wave * CHUNK;
  const uint32_t xs_base = (uint32_t)(uintptr_t)xs;
  for (int i = 0; i < CHUNK; ++i) {
    const int t = t0 + i;
    const uint32_t loff = xs_base + (uint32_t)((t * LSTRIDE + lane) * 4);
    const uint64_t gaddr =
        (uint64_t)(uintptr_t)(x + (size_t)t * BF + (size_t)col);
    async_ld_b32_to_lds(loff, gaddr);
  }
  wait_asynccnt0();
  __syncthreads();

  const float v  = v_init[col];
  const float x0 = xs[0 * LSTRIDE + lane];
  const float x1 = xs[1 * LSTRIDE + lane];
  const float h0 = 0.5f * (v + x0);   // v + (x0 - v)/TAU
  const float h1 = 0.5f * (h0 + x1);  // h0 + (x1 - h0)/TAU

  float* out_spike = out;
  float* out_y     = out + (size_t)T_DIM * BF;

  float y_reg[CHUNK];   // this wave's y chunk, carried across iterations

  // MAX_ITER = 10: first iteration uses the analytic warmstart guess,
  // last iteration fuses the spike/y global writes.
  lif_iteration<true,  false>(xs, summ, ybound, y_reg, wave, lane, v, h0, h1,
                              out_spike, out_y, col);
  for (int it = 0; it < 8; ++it) {
    lif_iteration<false, false>(xs, summ, ybound, y_reg, wave, lane, v, h0, h1,
                                out_spike, out_y, col);
  }
  lif_iteration<false, true >(xs, summ, ybound, y_reg, wave, lane, v, h0, h1,
                              out_spike, out_y, col);
}

// ---------------------------------------------------------------------------
extern "C" void kernel_launch(void* const* d_in, const int* in_sizes, int n_in,
                              void* d_out, int out_size, void* d_ws,
                              size_t ws_size, hipStream_t stream)
{
  const float* x      = (const float*)d_in[0];   // (T, B, F) f32
  const float* v_init = (const float*)d_in[1];   // (B, F)    f32
  float* out          = (float*)d_out;           // spike ++ y, each T*B*F f32

  const size_t shmem = (size_t)T_DIM * LSTRIDE * sizeof(float); // 135168 B
  hipFuncSetAttribute((const void*)deer_lif_kernel,
                      hipFuncAttributeMaxDynamicSharedMemorySize, (int)shmem);

  dim3 grid(BF / COLS);          // 512 blocks, each owns 32 columns fully
  dim3 block(NWAVES * 32);       // 1024 threads = 32 wave32
  deer_lif_kernel<<<grid, block, shmem, stream>>>(x, v_init, out);

  (void)in_sizes; (void)n_in; (void)out_size; (void)d_ws; (void)ws_size;
}